// LastAggregator_11416023072995
// MI455X (gfx1250) — compile-verified
//
#include <hip/hip_runtime.h>

// LastAggregator: segmented argmax-by-(t, pos) then row gather.
// Phase 1: zero 64-bit key workspace (dim_size entries).
// Phase 2: atomicMax of key = ((t+1) << 32) | pos  per event.
// Phase 3: one wave32 per segment copies the winning 128-float row (32 lanes x float4).

typedef unsigned long long u64;
typedef float v4f __attribute__((ext_vector_type(4)));
typedef int   v4i __attribute__((ext_vector_type(4)));

__global__ void lastagg_init_ws(u64* __restrict__ ws, int dim) {
  int i = blockIdx.x * blockDim.x + threadIdx.x;
  if (i < dim) ws[i] = 0ULL;
}

__global__ void lastagg_scatter(const int* __restrict__ index,
                                const int* __restrict__ t,
                                u64* __restrict__ ws, int n) {
  int base = (blockIdx.x * blockDim.x + threadIdx.x) * 4;
  if (base + 3 < n) {
    // b128 loads of 4 indices + 4 timestamps (streaming, non-temporal)
    v4i idx4 = __builtin_nontemporal_load((const v4i*)(index + base));
    v4i t4   = __builtin_nontemporal_load((const v4i*)(t + base));
#pragma unroll
    for (int k = 0; k < 4; ++k) {
      u64 key = ((u64)(unsigned)(t4[k] + 1) << 32) | (unsigned)(base + k);
      atomicMax(&ws[idx4[k]], key);   // -> global_atomic_max_u64, L2-resident
    }
  } else {
    for (int i = base; i < n; ++i) {
      u64 key = ((u64)(unsigned)(t[i] + 1) << 32) | (unsigned)i;
      atomicMax(&ws[index[i]], key);
    }
  }
}

__global__ void lastagg_gather(const v4f* __restrict__ msg,
                               const u64* __restrict__ ws,
                               v4f* __restrict__ out,
                               int dim, int nvec4 /* = D/4 */) {
  // One wave32 per segment. For D=128, nvec4=32: each lane moves exactly one
  // float4 -> one global_load_b128 + one global_store_b128 per lane per row.
  int wavesPerBlock = blockDim.x >> 5;
  int seg  = blockIdx.x * wavesPerBlock + (threadIdx.x >> 5);
  int lane = threadIdx.x & 31;
  if (seg >= dim) return;

  u64 key = ws[seg];                       // broadcast read (all lanes same addr)
  size_t obase = (size_t)seg * (size_t)nvec4;

  if (key != 0ULL) {
    size_t ibase = (size_t)(unsigned)(key & 0xFFFFFFFFu) * (size_t)nvec4;
    for (int v = lane; v < nvec4; v += 32) {
      v4f val = __builtin_nontemporal_load(msg + ibase + v);  // row read once
      __builtin_nontemporal_store(val, out + obase + v);      // out written once
    }
  } else {
    v4f z = (v4f){0.f, 0.f, 0.f, 0.f};
    for (int v = lane; v < nvec4; v += 32)
      __builtin_nontemporal_store(z, out + obase + v);
  }
}

extern "C" void kernel_launch(void* const* d_in, const int* in_sizes, int n_in,
                              void* d_out, int out_size, void* d_ws, size_t ws_size,
                              hipStream_t stream) {
  const float* msg   = (const float*)d_in[0];
  const int*   index = (const int*)d_in[1];
  const int*   t     = (const int*)d_in[2];
  // d_in[3] is dim_size as a device scalar; derive host-side values instead
  // (graph capture forbids sync copies):
  const int n   = in_sizes[1];              // number of events
  const int D   = in_sizes[0] / in_sizes[1];// feature dim (128)
  const int dim = out_size / D;             // number of segments (100000)

  u64* ws = (u64*)d_ws;                     // needs dim * 8 bytes (800 KB)

  // Phase 1: clear keys
  {
    int blocks = (dim + 255) / 256;
    lastagg_init_ws<<<blocks, 256, 0, stream>>>(ws, dim);
  }
  // Phase 2: fused (t, pos) argmax via 64-bit atomicMax
  {
    int threads = (n + 3) / 4;
    int blocks  = (threads + 255) / 256;
    lastagg_scatter<<<blocks, 256, 0, stream>>>(index, t, ws, n);
  }
  // Phase 3: wave-per-segment row gather (or zero fill)
  {
    const int wavesPerBlock = 8;            // 256 threads = 8 wave32
    int blocks = (dim + wavesPerBlock - 1) / wavesPerBlock;
    lastagg_gather<<<blocks, 256, 0, stream>>>((const v4f*)msg, ws,
                                               (v4f*)d_out, dim, D / 4);
  }
}